// GraphClassifier_7249904795690
// MI455X (gfx1250) — compile-verified
//
#include <hip/hip_runtime.h>
#include <math.h>

#define N_NODES 131072
#define N_EDGES 2097152
#define IN_DIM  768
#define HID     64
#define NPG     32
#define OUT_DIM 2

typedef float v2f __attribute__((ext_vector_type(2)));
typedef float v8f __attribute__((ext_vector_type(8)));

// ---------------- degree / normalization ----------------
__global__ void init_deg_kernel(float* __restrict__ deg) {
    int i = blockIdx.x * blockDim.x + threadIdx.x;
    deg[i] = 1.0f;  // self-loop
}

__global__ void deg_acc_kernel(const long long* __restrict__ ei,
                               float* __restrict__ deg) {
    int e = blockIdx.x * blockDim.x + threadIdx.x;
    int dst = (int)ei[N_EDGES + e];
    atomicAdd(&deg[dst], 1.0f);
}

__global__ void dinv_kernel(float* __restrict__ deg) {
    int i = blockIdx.x * blockDim.x + threadIdx.x;
    deg[i] = rsqrtf(deg[i]);
}

// ---------------- h = x @ W_conv via WMMA f32 16x16x4 ----------------
// Block: 256 threads = 8 wave32s. Each wave computes one 16x16 tile of h.
// Block tile: 32 rows x 64 cols. K tiled in 64-wide chunks staged in LDS.
__global__ __launch_bounds__(256) void gemm_kernel(const float* __restrict__ x,
                                                   const float* __restrict__ W,
                                                   float* __restrict__ h) {
    __shared__ float sW[64][65];  // [kk][n], pad to kill bank conflicts
    __shared__ float sX[32][65];  // [r][kk]

    const int t     = threadIdx.x;
    const int lane  = t & 31;
    const int wave  = t >> 5;
    const int rtile = wave >> 2;          // 0..1
    const int ctile = wave & 3;           // 0..3
    const int row0  = blockIdx.x * 32;
    const int m     = lane & 15;          // M (A) / N (B) within tile
    const int koff  = (lane >> 4) << 1;   // lanes 0-15: K+0/K+1, lanes 16-31: K+2/K+3

    v8f acc = {};

    for (int k0 = 0; k0 < IN_DIM; k0 += 64) {
        __syncthreads();
#pragma unroll
        for (int i = 0; i < 16; ++i) {    // 64x64 W chunk, coalesced
            int idx = t + i * 256;
            sW[idx >> 6][idx & 63] = W[(k0 + (idx >> 6)) * HID + (idx & 63)];
        }
#pragma unroll
        for (int i = 0; i < 8; ++i) {     // 32x64 x chunk, coalesced
            int idx = t + i * 256;
            sX[idx >> 6][idx & 63] = x[(size_t)(row0 + (idx >> 6)) * IN_DIM + k0 + (idx & 63)];
        }
        __syncthreads();

#pragma unroll
        for (int kk = 0; kk < 64; kk += 4) {
            v2f a, b;
            a.x = sX[rtile * 16 + m][kk + koff];
            a.y = sX[rtile * 16 + m][kk + koff + 1];
            b.x = sW[kk + koff][ctile * 16 + m];
            b.y = sW[kk + koff + 1][ctile * 16 + m];
            acc = __builtin_amdgcn_wmma_f32_16x16x4_f32(
                false, a, false, b, (short)0, acc, false, false);
        }
    }

    // D layout: lane<16 -> M=v, N=lane; lane>=16 -> M=v+8, N=lane-16
#pragma unroll
    for (int v = 0; v < 8; ++v) {
        int row = row0 + rtile * 16 + ((lane < 16) ? v : v + 8);
        int col = ctile * 16 + m;
        h[(size_t)row * HID + col] = acc[v];
    }
}

// ---------------- self-loop term initializes output features ----------------
__global__ void selfloop_kernel(const float* __restrict__ h,
                                const float* __restrict__ dinv,
                                float* __restrict__ outf) {
    int i = blockIdx.x * blockDim.x + threadIdx.x;
    int n = i >> 6;
    float d = dinv[n];
    outf[i] = h[i] * d * d;
}

// ---------------- edge gather + scaled scatter-add (L2-resident) ----------------
__global__ void edge_kernel(const long long* __restrict__ ei,
                            const float* __restrict__ h,
                            const float* __restrict__ dinv,
                            float* __restrict__ outf) {
    long long g = (long long)blockIdx.x * blockDim.x + threadIdx.x;
    int e = (int)(g >> 4);
    int c = ((int)g & 15) << 2;
    int src = (int)ei[e];
    int dst = (int)ei[N_EDGES + e];
    float nrm = dinv[src] * dinv[dst];
    const float4 hv = *(const float4*)(h + (size_t)src * HID + c);
    float* o = outf + (size_t)dst * HID + c;
    atomicAdd(o + 0, hv.x * nrm);
    atomicAdd(o + 1, hv.y * nrm);
    atomicAdd(o + 2, hv.z * nrm);
    atomicAdd(o + 3, hv.w * nrm);
}

// ---------------- bias + ReLU + [1x2048]@[2048x2] + log_softmax ----------------
__global__ __launch_bounds__(256) void head_kernel(const float* __restrict__ outf,
                                                   const float* __restrict__ b_conv,
                                                   const float* __restrict__ W_lin,
                                                   const float* __restrict__ b_lin,
                                                   float* __restrict__ out) {
    __shared__ float r0[256];
    __shared__ float r1[256];
    const int g = blockIdx.x;
    const int t = threadIdx.x;
    const float* f = outf + (size_t)g * (HID * NPG);
    float s0 = 0.f, s1 = 0.f;
    for (int i = t; i < HID * NPG; i += 256) {
        float v = fmaxf(f[i] + b_conv[i & (HID - 1)], 0.0f);
        s0 += v * W_lin[i];
        s1 += v * W_lin[HID * NPG + i];
    }
    r0[t] = s0; r1[t] = s1;
    __syncthreads();
    for (int ofs = 128; ofs > 0; ofs >>= 1) {
        if (t < ofs) { r0[t] += r0[t + ofs]; r1[t] += r1[t + ofs]; }
        __syncthreads();
    }
    if (t == 0) {
        float l0 = r0[0] + b_lin[0];
        float l1 = r1[0] + b_lin[1];
        float mx = fmaxf(l0, l1);
        float lse = mx + logf(expf(l0 - mx) + expf(l1 - mx));
        out[g * 2 + 0] = l0 - lse;
        out[g * 2 + 1] = l1 - lse;
    }
}

// ---------------- launcher ----------------
extern "C" void kernel_launch(void* const* d_in, const int* in_sizes, int n_in,
                              void* d_out, int out_size, void* d_ws, size_t ws_size,
                              hipStream_t stream) {
    const float*     x  = (const float*)d_in[0];
    const long long* ei = (const long long*)d_in[1];   // int64 edge_index [2, E]
    const float*     Wc = (const float*)d_in[2];       // [768, 64] row-major
    const float*     bc = (const float*)d_in[3];       // [64]
    const float*     Wl = (const float*)d_in[4];       // [2, 2048] row-major
    const float*     bl = (const float*)d_in[5];       // [2]
    float* out = (float*)d_out;                        // [4096, 2]

    float* ws   = (float*)d_ws;
    float* deg  = ws;                                   // N floats (becomes dinv)
    float* h    = ws + N_NODES;                         // N*HID floats
    float* outf = h + (size_t)N_NODES * HID;            // N*HID floats

    init_deg_kernel<<<N_NODES / 256, 256, 0, stream>>>(deg);
    deg_acc_kernel<<<N_EDGES / 256, 256, 0, stream>>>(ei, deg);
    dinv_kernel<<<N_NODES / 256, 256, 0, stream>>>(deg);
    gemm_kernel<<<N_NODES / 32, 256, 0, stream>>>(x, Wc, h);
    selfloop_kernel<<<(N_NODES * HID) / 256, 256, 0, stream>>>(h, deg, outf);
    edge_kernel<<<(N_EDGES / 256) * 16, 256, 0, stream>>>(ei, h, deg, outf);
    head_kernel<<<N_NODES / NPG, 256, 0, stream>>>(outf, bc, Wl, bl, out);
}